// TopKGate_44856638439904
// MI455X (gfx1250) — compile-verified
//
#include <hip/hip_runtime.h>
#include <math.h>

#define S   8192
#define Dd  2048
#define E   64
#define CAP 128

typedef __attribute__((ext_vector_type(2))) float v2f;
typedef __attribute__((ext_vector_type(4))) float v4f;
typedef __attribute__((ext_vector_type(8))) float v8f;

// ---------------------------------------------------------------------------
// Kernel 0: streaming zero-fill of d_out (537 MB -> the real bottleneck).
// Non-temporal b128 stores: output is 537MB > 192MB L2, pure write stream.
// ---------------------------------------------------------------------------
__global__ __launch_bounds__(256) void tg_zero(v4f* __restrict__ p, long long nvec,
                                               float* __restrict__ tail, int ntail) {
    long long i = (long long)blockIdx.x * blockDim.x + threadIdx.x;
    const long long stride = (long long)gridDim.x * blockDim.x;
    v4f z = {0.f, 0.f, 0.f, 0.f};
    for (; i < nvec; i += stride)
        __builtin_nontemporal_store(z, &p[i]);
    if (blockIdx.x == 0 && threadIdx.x == 0)
        for (int t = 0; t < ntail; ++t) tail[t] = 0.0f;
}

// ---------------------------------------------------------------------------
// Kernel 1: logits = x @ wg via V_WMMA_F32_16X16X4_F32, then per-row softmax
// + argmax. One wave (32 threads) per 16-token tile; 4 accumulators cover
// all 64 experts. A-frag: lane L<16 holds row (L), K=k..k+1; lanes 16-31 hold
// K=k+2..k+3 (ISA 16x4 f32 A layout). B-frag analogous (4x16).
// C/D layout: VGPR j = row j (lanes 0-15) / row 8+j (lanes 16-31), col = L%16.
// ---------------------------------------------------------------------------
__global__ __launch_bounds__(32) void tg_gemm_softmax(
        const float* __restrict__ x, const float* __restrict__ wg,
        float* __restrict__ gates, int* __restrict__ idx_out,
        float* __restrict__ gval) {
    const int lane = threadIdx.x;
    const int half = lane >> 4;       // 0: K lo pair, 1: K hi pair
    const int l16  = lane & 15;
    const int r0   = blockIdx.x * 16;

    v8f acc0 = {}, acc1 = {}, acc2 = {}, acc3 = {};
    const float* arow = x + (size_t)(r0 + l16) * Dd + half * 2;

    for (int k = 0; k < Dd; k += 4) {
        v2f a = *(const v2f*)(arow + k);
        const float* bp = wg + (size_t)(k + half * 2) * E + l16;
        v2f b0, b1, b2, b3;
        b0.x = bp[0];  b0.y = bp[E +  0];
        b1.x = bp[16]; b1.y = bp[E + 16];
        b2.x = bp[32]; b2.y = bp[E + 32];
        b3.x = bp[48]; b3.y = bp[E + 48];
        acc0 = __builtin_amdgcn_wmma_f32_16x16x4_f32(false, a, false, b0, (short)0, acc0, false, false);
        acc1 = __builtin_amdgcn_wmma_f32_16x16x4_f32(false, a, false, b1, (short)0, acc1, false, false);
        acc2 = __builtin_amdgcn_wmma_f32_16x16x4_f32(false, a, false, b2, (short)0, acc2, false, false);
        acc3 = __builtin_amdgcn_wmma_f32_16x16x4_f32(false, a, false, b3, (short)0, acc3, false, false);
    }

    __shared__ float lds[16][E + 1];   // +1 pad: conflict-free row access
    #pragma unroll
    for (int j = 0; j < 8; ++j) {
        const int row = half * 8 + j;
        lds[row][ 0 + l16] = acc0[j];
        lds[row][16 + l16] = acc1[j];
        lds[row][32 + l16] = acc2[j];
        lds[row][48 + l16] = acc3[j];
    }
    __syncthreads();

    if (lane < 16) {
        const int s = r0 + lane;
        // argmax (first-max tie-break, matching jnp.argmax) + max for softmax
        float m = lds[lane][0]; int am = 0;
        for (int e = 1; e < E; ++e) {
            float v = lds[lane][e];
            if (v > m) { m = v; am = e; }
        }
        float sum = 0.f;
        for (int e = 0; e < E; ++e) {
            float ex = expf(lds[lane][e] - m);
            lds[lane][e] = ex;
            sum += ex;
        }
        const float inv = 1.0f / sum;
        float* grow = gates + (size_t)s * E;
        for (int e = 0; e < E; ++e) grow[e] = lds[lane][e] * inv;
        idx_out[s] = am;
        gval[s]    = lds[lane][am] * inv;
    }
}

// ---------------------------------------------------------------------------
// Kernel 2: me[e] = mean_s gates[s,e]  (deterministic tree reduction)
// ---------------------------------------------------------------------------
__global__ __launch_bounds__(256) void tg_me(const float* __restrict__ gates,
                                             float* __restrict__ me) {
    __shared__ float red[256];
    const int e = blockIdx.x, t = threadIdx.x;
    float sum = 0.f;
    for (int s = t; s < S; s += 256) sum += gates[(size_t)s * E + e];
    red[t] = sum; __syncthreads();
    for (int w = 128; w > 0; w >>= 1) {
        if (t < w) red[t] += red[t + w];
        __syncthreads();
    }
    if (t == 0) me[e] = red[0] * (1.0f / (float)S);
}

// ---------------------------------------------------------------------------
// Kernel 3: deterministic capacity scan. 64 owner-threads each scan the token
// stream (staged in LDS) for their expert, assigning in-order ranks; first
// CAP tokens kept. ce uses pre-drop counts (as in reference). Finishes l_aux
// with a fixed-order sum -> bit-identical across graph replays.
// ---------------------------------------------------------------------------
__global__ __launch_bounds__(256) void tg_scan(const int* __restrict__ idx,
                                               const float* __restrict__ me,
                                               int* __restrict__ off,
                                               int* __restrict__ keep,
                                               float* __restrict__ laux_out) {
    __shared__ int   chunk[2048];
    __shared__ float part[E];
    const int t = threadIdx.x;
    int c = 0;
    for (int base = 0; base < S; base += 2048) {
        for (int i = t; i < 2048; i += 256) chunk[i] = idx[base + i];
        __syncthreads();
        if (t < E) {
            for (int i = 0; i < 2048; ++i) {
                if (chunk[i] == t) {
                    const int s = base + i;
                    if (c < CAP) { off[s] = c; keep[s] = 1; }
                    else         { off[s] = 0; keep[s] = 0; }
                    ++c;
                }
            }
        }
        __syncthreads();
    }
    if (t < E) part[t] = me[t] * ((float)c / (float)S);
    __syncthreads();
    if (t == 0) {
        float l = 0.f;
        for (int e = 0; e < E; ++e) l += part[e];
        laux_out[0] = l * (float)E;
    }
}

// ---------------------------------------------------------------------------
// Kernel 4: sparse scatter — one nonzero (e,c) cell per kept token.
// ---------------------------------------------------------------------------
__global__ __launch_bounds__(256) void tg_scatter(const int* __restrict__ idx,
                                                  const int* __restrict__ keep,
                                                  const int* __restrict__ off,
                                                  const float* __restrict__ gval,
                                                  float* __restrict__ out) {
    const int s = blockIdx.x * 256 + threadIdx.x;
    if (s >= S) return;
    if (!keep[s]) return;
    const long long pos = ((long long)s * E + idx[s]) * CAP + off[s];
    float* cw = out + 1;                               // combine_weights
    cw[pos] = gval[s];
    cw[(long long)S * E * CAP + pos] = 1.0f;           // dispatch_mask (>0)
}

// ---------------------------------------------------------------------------
extern "C" void kernel_launch(void* const* d_in, const int* in_sizes, int n_in,
                              void* d_out, int out_size, void* d_ws, size_t ws_size,
                              hipStream_t stream) {
    const float* x  = (const float*)d_in[0];
    const float* wg = (const float*)d_in[1];
    float* out = (float*)d_out;

    // workspace layout
    float* gates = (float*)d_ws;                 // S*E floats (2 MB)
    float* me    = gates + (size_t)S * E;        // 64 floats
    int*   idx   = (int*)(me + E);               // S ints
    float* gval  = (float*)(idx + S);            // S floats
    int*   off   = (int*)(gval + S);             // S ints
    int*   keep  = off + S;                      // S ints

    // 0) zero 537MB output (the bandwidth-dominant step), NT b128 stores
    const long long nvec  = (long long)out_size / 4;
    const int       ntail = (int)((long long)out_size - nvec * 4);
    tg_zero<<<4096, 256, 0, stream>>>((v4f*)out, nvec, out + nvec * 4, ntail);

    // 1) WMMA f32 GEMM + softmax + argmax (512 waves, one 16-token tile each)
    tg_gemm_softmax<<<S / 16, 32, 0, stream>>>(x, wg, gates, idx, gval);

    // 2) me[e] column means (deterministic)
    tg_me<<<E, 256, 0, stream>>>(gates, me);

    // 3) capacity scan + l_aux -> d_out[0]
    tg_scan<<<1, 256, 0, stream>>>(idx, me, off, keep, out);

    // 4) scatter the 8192 nonzero cells
    tg_scatter<<<(S + 255) / 256, 256, 0, stream>>>(idx, keep, off, gval, out);
}